// FunctionAffineGridBilinearSample_54039278519070
// MI455X (gfx1250) — compile-verified
//
#include <hip/hip_runtime.h>

// Bilinear grid sample (align_corners=True, zero padding), f32.
// src: [B, C, H, W], grid: [B, Ho, Wo, 2] (x,y in [-1,1]), out: [B, C, Ho, Wo]
// B=8, C=128, H=W=Ho=Wo=128.
//
// Memory-bound gather: ~136 MB of compulsory HBM traffic -> ~5.8 us floor at
// 23.3 TB/s. src (64 MiB) is kept L2-resident (RT loads); the streaming output
// is written with non-temporal stores so it does not evict src from the 192 MB
// L2. No WMMA: the 4-tap reduction has per-pixel gather indices (no shared K),
// so matrix ops are inapplicable by construction.

namespace {
constexpr int B  = 8;
constexpr int C  = 128;
constexpr int H  = 128;
constexpr int W  = 128;
constexpr int Ho = 128;
constexpr int Wo = 128;
constexpr int HW = H * W;            // 16384 elements = 64 KiB per channel plane
constexpr int CHW = C * HW;          // per-batch stride (src and out identical here)
constexpr int NPIX = B * Ho * Wo;    // 131072 output pixels
}

__global__ __launch_bounds__(256)
void grid_sample_bilinear_f32(const float* __restrict__ src,
                              const float* __restrict__ grid,
                              float* __restrict__ out) {
    const int pix = blockIdx.x * 256 + threadIdx.x;
    if (pix >= NPIX) return;

    const int wo = pix & (Wo - 1);
    const int ho = (pix >> 7) & (Ho - 1);
    const int b  = pix >> 14;

    // Coalesced 8-byte grid read: lane = consecutive pixel -> consecutive float2.
    const float2 g = reinterpret_cast<const float2*>(grid)[pix];

    // align_corners=True unnormalization
    const float x = (g.x + 1.0f) * (0.5f * (float)(W - 1));
    const float y = (g.y + 1.0f) * (0.5f * (float)(H - 1));

    const float x0f = floorf(x);
    const float y0f = floorf(y);
    const float wx1 = x - x0f;
    const float wy1 = y - y0f;
    const float wx0 = 1.0f - wx1;
    const float wy0 = 1.0f - wy1;

    int x0 = (int)x0f, y0 = (int)y0f;
    int x1 = x0 + 1,   y1 = y0 + 1;

    // Zero-padding masks folded into the weights (matches reference).
    const float mx0 = (x0 >= 0 && x0 < W) ? 1.0f : 0.0f;
    const float mx1 = (x1 >= 0 && x1 < W) ? 1.0f : 0.0f;
    const float my0 = (y0 >= 0 && y0 < H) ? 1.0f : 0.0f;
    const float my1 = (y1 >= 0 && y1 < H) ? 1.0f : 0.0f;

    const float w00 = (wy0 * wx0) * (my0 * mx0);
    const float w01 = (wy0 * wx1) * (my0 * mx1);
    const float w10 = (wy1 * wx0) * (my1 * mx0);
    const float w11 = (wy1 * wx1) * (my1 * mx1);

    // Clamp for safe addressing (masked taps contribute 0 anyway).
    x0 = min(max(x0, 0), W - 1);  x1 = min(max(x1, 0), W - 1);
    y0 = min(max(y0, 0), H - 1);  y1 = min(max(y1, 0), H - 1);

    const float* __restrict__ sb = src + (size_t)b * CHW;
    const int o00 = y0 * W + x0;
    const int o01 = y0 * W + x1;   // same cache line as o00 unless x crosses a 32-float boundary
    const int o10 = y1 * W + x0;
    const int o11 = y1 * W + x1;

    // Warm L2 with this thread's two gather lines while the ALU work above
    // retires (gfx1250 global_prefetch_b8 path).
    __builtin_prefetch(sb + o00, 0, 3);
    __builtin_prefetch(sb + o10, 0, 3);

    float* __restrict__ ob = out + (size_t)b * CHW + ho * Wo + wo;

    // Inner loop over channels: 4 independent gathers per iteration; unroll x8
    // -> 32 loads in flight per lane. Channel offsets (c * 64 KiB, c < 128)
    // fit the signed 24-bit immediate offset, so unrolled iterations share one
    // address VGPR. Stores are fully coalesced across the wave (consecutive wo)
    // and non-temporal so the 64 MiB output stream doesn't evict src from L2.
    #pragma unroll 8
    for (int c = 0; c < C; ++c) {
        const int coff = c * HW;
        const float v00 = sb[coff + o00];
        const float v01 = sb[coff + o01];
        const float v10 = sb[coff + o10];
        const float v11 = sb[coff + o11];
        const float r = fmaf(v00, w00,
                        fmaf(v01, w01,
                        fmaf(v10, w10, v11 * w11)));
        __builtin_nontemporal_store(r, ob + (size_t)c * HW);
    }
}

extern "C" void kernel_launch(void* const* d_in, const int* in_sizes, int n_in,
                              void* d_out, int out_size, void* d_ws, size_t ws_size,
                              hipStream_t stream) {
    (void)in_sizes; (void)n_in; (void)out_size; (void)d_ws; (void)ws_size;
    const float* src  = (const float*)d_in[0];
    const float* grid = (const float*)d_in[1];
    float* out = (float*)d_out;

    const int threads = 256;
    const int blocks  = (NPIX + threads - 1) / threads;  // 512
    grid_sample_bilinear_f32<<<blocks, threads, 0, stream>>>(src, grid, out);
}